// CostVolume_23081154249136
// MI455X (gfx1250) — compile-verified
//
#include <hip/hip_runtime.h>

typedef __attribute__((ext_vector_type(2))) float v2f;
typedef __attribute__((ext_vector_type(8))) float v8f;

namespace {
constexpr int B = 8, C = 128, H = 160, W = 160;
constexpr int HW = H * W;
constexpr long CHW = (long)C * HW;
constexpr int R = 2, NOFF = 5;                       // 5x5 search window
constexpr int TILE_OUT = 12;                         // owned pixels per 16-wide WMMA tile
constexpr int TPR = (W + TILE_OUT - 1) / TILE_OUT;   // 14 tiles per image row
constexpr int WAVES_PER_BLOCK = 8;                   // 256 threads, wave32
constexpr int TOTAL_TILES = B * H * TPR;             // 17920 (divisible by 8)
}

__device__ __forceinline__ int refl(int i, int n) {
  i = (i < 0) ? -i : i;              // reflect low edge (r=2 < n, single bounce)
  i = (i >= n) ? (2 * n - 2 - i) : i; // reflect high edge
  return i;
}

__global__ __launch_bounds__(256) void corr_wmma_kernel(
    const float* __restrict__ c1, const float* __restrict__ c2,
    const float* __restrict__ prelu, float* __restrict__ out) {
  const int lane = threadIdx.x & 31;
  const int wave = threadIdx.x >> 5;
  const int tile = blockIdx.x * WAVES_PER_BLOCK + wave;

  const int t  = tile % TPR;
  const int h  = (tile / TPR) % H;
  const int b  = tile / (TPR * H);
  const int w0 = t * TILE_OUT;

  const int half = lane >> 4;   // 0: lanes 0-15 (K 0,1 / M 0-7), 1: lanes 16-31 (K 2,3 / M 8-15)
  const int l16  = lane & 15;

  // A-matrix row (M): c1 pixel w0+l16, clamped for the unowned tail rows.
  const int aw = (w0 + l16 < W) ? (w0 + l16) : (W - 1);
  // B-matrix column (N): c2 search-window column w0-2+l16, reflect-padded.
  const int bw = refl(w0 - 2 + l16, W);

  int hs[NOFF];
#pragma unroll
  for (int dy = 0; dy < NOFF; ++dy) hs[dy] = refl(h + dy - R, H);

  const float* c1p = c1 + (long)b * CHW + (long)h * W + aw;
  const float* c2p[NOFF];
#pragma unroll
  for (int dy = 0; dy < NOFF; ++dy)
    c2p[dy] = c2 + (long)b * CHW + (long)hs[dy] * W + bw;

  v8f acc[NOFF] = {};  // five 16x16 f32 accumulators (one per dy), 40 VGPRs

  // K loop over channels, 4 per V_WMMA_F32_16X16X4_F32. A loaded once,
  // reused by all 5 dy WMMAs. No divergence in this loop (EXEC all-1s).
  for (int k0 = 0; k0 < C; k0 += 4) {
    const long co = (long)(k0 + 2 * half) * HW;
    v2f a;
    a.x = c1p[co];
    a.y = c1p[co + HW];
#pragma unroll
    for (int dy = 0; dy < NOFF; ++dy) {
      v2f bm;
      bm.x = c2p[dy][co];
      bm.y = c2p[dy][co + HW];
      acc[dy] = __builtin_amdgcn_wmma_f32_16x16x4_f32(
          false, a, false, bm, (short)0, acc[dy], false, false);
    }
  }

  // Epilogue: D[i][j] is useful when dx = j - i is in [0,4]. Lane holds
  // column j = l16; VGPR v holds row i = v + 8*half. Scale, PReLU, scatter.
  const float alpha = prelu[0];
  const float inv_c = 1.0f / (float)C;
  float* outb = out + (long)b * (NOFF * NOFF) * HW + (long)h * W;
#pragma unroll
  for (int dy = 0; dy < NOFF; ++dy) {
#pragma unroll
    for (int v = 0; v < 8; ++v) {
      const int i  = v + 8 * half;   // pixel index within tile
      const int dx = l16 - i;        // x-offset index 0..4
      const int wp = w0 + i;
      if (dx >= 0 && dx < NOFF && i < TILE_OUT && wp < W) {
        float val = acc[dy][v] * inv_c;
        val = (val >= 0.0f) ? val : alpha * val;
        outb[(long)(dy * NOFF + dx) * HW + wp] = val;
      }
    }
  }
}

extern "C" void kernel_launch(void* const* d_in, const int* in_sizes, int n_in,
                              void* d_out, int out_size, void* d_ws, size_t ws_size,
                              hipStream_t stream) {
  (void)in_sizes; (void)n_in; (void)out_size; (void)d_ws; (void)ws_size;
  const float* c1    = (const float*)d_in[0];
  const float* c2    = (const float*)d_in[1];
  const float* prelu = (const float*)d_in[2];
  float* out = (float*)d_out;

  dim3 block(32 * WAVES_PER_BLOCK);                 // 256 threads = 8 wave32
  dim3 grid(TOTAL_TILES / WAVES_PER_BLOCK);         // 2240 blocks
  corr_wmma_kernel<<<grid, block, 0, stream>>>(c1, c2, prelu, out);
}